// GIN_MUTAG_66116726554993
// MI455X (gfx1250) — compile-verified
//
#include <hip/hip_runtime.h>

typedef __attribute__((ext_vector_type(2))) float v2f;
typedef __attribute__((ext_vector_type(8))) float v8f;

#define HDIM 20
#define NCLS 2
#define BN_EPS 1e-5f

// ---------------------------------------------------------------------------
// Copy h into z buffer with K padded to a multiple of 4 (zero fill), since
// z = h + agg and the WMMA kernel wants guard-free K-aligned A loads.
// ---------------------------------------------------------------------------
__global__ void copy_pad_kernel(const float* __restrict__ src, float* __restrict__ dst,
                                int nrows, int din, int zs) {
  int idx = blockIdx.x * blockDim.x + threadIdx.x;
  if (idx >= nrows * zs) return;
  int row = idx / zs;
  int j = idx - row * zs;
  dst[idx] = (j < din) ? src[(long)row * din + j] : 0.0f;
}

// ---------------------------------------------------------------------------
// Weighted scatter-add over edges: agg[dst] += h[src] * w   (fp32 atomics, L2)
// agg has stride zs (padded); h has its natural stride din.
// ---------------------------------------------------------------------------
__global__ void edge_scatter_kernel(const int* __restrict__ esrc, const int* __restrict__ edst,
                                    const float* __restrict__ ew, const float* __restrict__ h,
                                    float* __restrict__ agg, int E, int din, int zs) {
  int e = blockIdx.x * blockDim.x + threadIdx.x;
  if (e >= E) return;
  int s = esrc[e];
  int t = edst[e];
  float w = ew[e];
  const float* hs = h + (long)s * din;
  float* at = agg + (long)t * zs;
  __builtin_prefetch(at, 1, 1);   // global_prefetch_b8: warm the atomic target line
  #pragma unroll 4
  for (int j = 0; j < din; ++j) atomicAdd(&at[j], hs[j] * w);
}

// ---------------------------------------------------------------------------
// Fused MLP:  u = relu(z @ W1 + b1) @ W2 + b2
// One wave per 16-row tile; V_WMMA_F32_16X16X4_F32 (fp32 WMMA, wave32).
// DIN is compile-time; K padded to KP (mult of 4). Weights staged in LDS,
// transposed ([col][k], k contiguous) and zero-padded to 32 cols, so every
// fragment load is an unconditional aligned 8-byte load. No predication in
// the WMMA loops -> EXEC stays all-ones throughout.
// ---------------------------------------------------------------------------
template <int DIN, int ZS>
__global__ void __launch_bounds__(256)
mlp_wmma_kernel(const float* __restrict__ z, const float* __restrict__ W1,
                const float* __restrict__ b1, const float* __restrict__ W2,
                const float* __restrict__ b2, float* __restrict__ u, int nrows) {
  constexpr int KP1 = (DIN + 3) & ~3;      // padded K of GEMM1 (8 or 20)
  __shared__ float w1t[32][KP1];           // W1 transposed, cols padded to 32
  __shared__ float w2t[32][HDIM];          // W2 transposed, cols padded to 32
  __shared__ float lds_t[8][16][24];       // per-wave relu(z@W1+b1) tile

  // ---- stage weights (transposed, zero-padded) ----
  for (int i = threadIdx.x; i < 32 * KP1; i += blockDim.x) {
    int c = i / KP1, k = i - c * KP1;
    w1t[c][k] = (c < HDIM && k < DIN) ? W1[k * HDIM + c] : 0.0f;
  }
  for (int i = threadIdx.x; i < 32 * HDIM; i += blockDim.x) {
    int c = i / HDIM, k = i - c * HDIM;
    w2t[c][k] = (c < HDIM) ? W2[k * HDIM + c] : 0.0f;
  }
  __syncthreads();

  const int wave = threadIdx.x >> 5;
  const int lane = threadIdx.x & 31;
  const int r    = lane & 15;              // row (A fragment) / col (B/D fragment)
  const int half = lane >> 4;              // selects K pair / row half
  const int rowbase = (blockIdx.x * 8 + wave) * 16;
  const int row  = rowbase + r;
  const int rowc = row < nrows ? row : nrows - 1;   // clamp: tail results discarded
  const int c1col = 16 + r;
  const bool c1ok = c1col < HDIM;

  // ---- GEMM 1: z(16 x KP1) @ W1(KP1 x 20) ----
  v8f acc0 = {0.f,0.f,0.f,0.f,0.f,0.f,0.f,0.f};
  v8f acc1 = {0.f,0.f,0.f,0.f,0.f,0.f,0.f,0.f};
  #pragma unroll
  for (int kt = 0; kt < KP1 / 4; ++kt) {
    const int k0 = kt * 4 + half * 2;      // even -> 8B aligned
    v2f a  = *(const v2f*)&z[(long)rowc * ZS + k0];
    v2f bA = *(const v2f*)&w1t[r][k0];
    v2f bB = *(const v2f*)&w1t[16 + r][k0];
    acc0 = __builtin_amdgcn_wmma_f32_16x16x4_f32(false, a, false, bA, (short)0, acc0, false, false);
    acc1 = __builtin_amdgcn_wmma_f32_16x16x4_f32(false, a, false, bB, (short)0, acc1, false, false);
  }

  // bias + ReLU -> LDS staging tile (D-fragment layout -> row-major)
  const float bias1a = b1[r];
  const float bias1b = c1ok ? b1[c1col] : 0.0f;
  #pragma unroll
  for (int i = 0; i < 8; ++i) {
    const int rr = i + half * 8;
    float v0 = acc0[i] + bias1a;
    lds_t[wave][rr][r] = v0 > 0.0f ? v0 : 0.0f;
    if (c1ok) {
      float v1 = acc1[i] + bias1b;
      lds_t[wave][rr][c1col] = v1 > 0.0f ? v1 : 0.0f;
    }
  }
  __syncthreads();

  // ---- GEMM 2: t(16 x 20) @ W2(20 x 20) ----
  v8f d0 = {0.f,0.f,0.f,0.f,0.f,0.f,0.f,0.f};
  v8f d1 = {0.f,0.f,0.f,0.f,0.f,0.f,0.f,0.f};
  #pragma unroll
  for (int kt = 0; kt < HDIM / 4; ++kt) {
    const int k0 = kt * 4 + half * 2;      // max 18; k0+1 = 19 < 20 always
    v2f a  = *(const v2f*)&lds_t[wave][r][k0];
    v2f bA = *(const v2f*)&w2t[r][k0];
    v2f bB = *(const v2f*)&w2t[16 + r][k0];
    d0 = __builtin_amdgcn_wmma_f32_16x16x4_f32(false, a, false, bA, (short)0, d0, false, false);
    d1 = __builtin_amdgcn_wmma_f32_16x16x4_f32(false, a, false, bB, (short)0, d1, false, false);
  }

  // bias + store u (pre-BN activations)
  const float bias2a = b2[r];
  const float bias2b = c1ok ? b2[c1col] : 0.0f;
  #pragma unroll
  for (int i = 0; i < 8; ++i) {
    const int rr = rowbase + i + half * 8;
    if (rr < nrows) {
      u[(long)rr * HDIM + r] = d0[i] + bias2a;
      if (c1ok) u[(long)rr * HDIM + c1col] = d1[i] + bias2b;
    }
  }
}

// ---------------------------------------------------------------------------
// BatchNorm statistics: one block per feature column.
// Emits fused scale = gamma * rsqrt(var+eps), shift = beta - mean*scale.
// ---------------------------------------------------------------------------
__global__ void bn_stats_kernel(const float* __restrict__ u, const float* __restrict__ gamma,
                                const float* __restrict__ beta, float* __restrict__ scale,
                                float* __restrict__ shift, int n) {
  const int j = blockIdx.x;  // feature column, 0..HDIM-1
  float s = 0.0f, ss = 0.0f;
  for (int i = threadIdx.x; i < n; i += blockDim.x) {
    float v = u[(long)i * HDIM + j];
    s += v;
    ss += v * v;
  }
  __shared__ float sh_s[256];
  __shared__ float sh_q[256];
  sh_s[threadIdx.x] = s;
  sh_q[threadIdx.x] = ss;
  __syncthreads();
  for (int o = 128; o > 0; o >>= 1) {
    if (threadIdx.x < o) {
      sh_s[threadIdx.x] += sh_s[threadIdx.x + o];
      sh_q[threadIdx.x] += sh_q[threadIdx.x + o];
    }
    __syncthreads();
  }
  if (threadIdx.x == 0) {
    float mean = sh_s[0] / (float)n;
    float var  = sh_q[0] / (float)n - mean * mean;
    float inv  = rsqrtf(var + BN_EPS);
    float sc   = inv * gamma[j];
    scale[j] = sc;
    shift[j] = beta[j] - mean * sc;
  }
}

// ---------------------------------------------------------------------------
// Fused BN + ReLU: h = relu(u*scale + shift)
// ---------------------------------------------------------------------------
__global__ void bn_apply_kernel(const float* __restrict__ u, const float* __restrict__ scale,
                                const float* __restrict__ shift, float* __restrict__ hout, int n) {
  int idx = blockIdx.x * blockDim.x + threadIdx.x;
  if (idx >= n) return;
  int j = idx % HDIM;
  float v = u[idx] * scale[j] + shift[j];
  hout[idx] = v > 0.0f ? v : 0.0f;
}

// ---------------------------------------------------------------------------
// Add-pool: graph_emb[batch[i]] += h[i]
// ---------------------------------------------------------------------------
__global__ void pool_kernel(const float* __restrict__ h, const int* __restrict__ batch,
                            float* __restrict__ ge, int n) {
  int i = blockIdx.x * blockDim.x + threadIdx.x;
  if (i >= n) return;
  int g = batch[i];
  const float* hi = h + (long)i * HDIM;
  float* gg = ge + (long)g * HDIM;
  #pragma unroll
  for (int j = 0; j < HDIM; ++j) atomicAdd(&gg[j], hi[j]);
}

// ---------------------------------------------------------------------------
// Final FC: out = graph_emb @ fc_W + fc_b    (G x 20) @ (20 x 2)
// ---------------------------------------------------------------------------
__global__ void fc_kernel(const float* __restrict__ ge, const float* __restrict__ W,
                          const float* __restrict__ b, float* __restrict__ out, int total) {
  int idx = blockIdx.x * blockDim.x + threadIdx.x;
  if (idx >= total) return;
  int g = idx / NCLS;
  int c = idx % NCLS;
  float acc = b[c];
  const float* row = ge + (long)g * HDIM;
  #pragma unroll
  for (int k = 0; k < HDIM; ++k) acc += row[k] * W[k * NCLS + c];
  out[idx] = acc;
}

// ---------------------------------------------------------------------------
extern "C" void kernel_launch(void* const* d_in, const int* in_sizes, int n_in,
                              void* d_out, int out_size, void* d_ws, size_t ws_size,
                              hipStream_t stream) {
  const float* x     = (const float*)d_in[0];
  const int*   eidx  = (const int*)d_in[1];
  const int*   batch = (const int*)d_in[2];
  const float* ew    = (const float*)d_in[3];

  const int N = in_sizes[2];            // 100000
  const int E = in_sizes[3];            // 3200000
  const int F = in_sizes[0] / N;        // 7
  const int G = (out_size - N * HDIM) / (HDIM + NCLS);  // 1000

  const float *l1W[3], *l1b[3], *l2W[3], *l2b[3], *gam[3], *bet[3];
  const float *fcW, *fcb;
  if (n_in >= 24) {
    // Lists flattened into separate inputs
    for (int l = 0; l < 3; ++l) {
      l1W[l] = (const float*)d_in[4 + l];
      l1b[l] = (const float*)d_in[7 + l];
      l2W[l] = (const float*)d_in[10 + l];
      l2b[l] = (const float*)d_in[13 + l];
      gam[l] = (const float*)d_in[16 + l];
      bet[l] = (const float*)d_in[19 + l];
    }
    fcW = (const float*)d_in[22];
    fcb = (const float*)d_in[23];
  } else {
    // Lists concatenated into one input each
    const float* w1 = (const float*)d_in[4];
    const float* b1 = (const float*)d_in[5];
    const float* w2 = (const float*)d_in[6];
    const float* b2 = (const float*)d_in[7];
    const float* gg = (const float*)d_in[8];
    const float* bb = (const float*)d_in[9];
    l1W[0] = w1;  l1W[1] = w1 + F * HDIM;  l1W[2] = w1 + F * HDIM + HDIM * HDIM;
    for (int l = 0; l < 3; ++l) {
      l1b[l] = b1 + HDIM * l;
      l2W[l] = w2 + HDIM * HDIM * l;
      l2b[l] = b2 + HDIM * l;
      gam[l] = gg + HDIM * l;
      bet[l] = bb + HDIM * l;
    }
    fcW = (const float*)d_in[10];
    fcb = (const float*)d_in[11];
  }

  const int* esrc = eidx;
  const int* edst = eidx + E;

  // Workspace layout (fp32)
  float* ws    = (float*)d_ws;
  float* zbuf  = ws;                         // N*HDIM (agg/z, K-padded stride)
  float* ubuf  = ws + (size_t)N * HDIM;      // N*HDIM (pre-BN activations)
  float* hbuf  = ws + (size_t)2 * N * HDIM;  // N*HDIM (layer output)
  float* scale = ws + (size_t)3 * N * HDIM;
  float* shift = scale + HDIM;

  // Output layout: out[G*C] | node_embs[N*H] | graph_emb[G*H]
  float* out_ptr   = (float*)d_out;
  float* node_embs = out_ptr + (size_t)G * NCLS;
  float* graph_emb = node_embs + (size_t)N * HDIM;

  // Zero output (graph_emb is atomically accumulated; out/node_embs overwritten)
  hipMemsetAsync(d_out, 0, (size_t)out_size * sizeof(float), stream);

  const int nblocks = (N + 127) / 128;  // 8 waves/block, 16 rows/wave

  const float* hcur = x;
  int dcur = F;
  for (int l = 0; l < 3; ++l) {
    const int zs = (dcur + 3) & ~3;     // padded stride: 8 (layer 0) or 20
    copy_pad_kernel<<<(N * zs + 255) / 256, 256, 0, stream>>>(hcur, zbuf, N, dcur, zs);
    edge_scatter_kernel<<<(E + 255) / 256, 256, 0, stream>>>(esrc, edst, ew, hcur, zbuf,
                                                             E, dcur, zs);
    if (dcur == HDIM) {
      mlp_wmma_kernel<HDIM, HDIM><<<nblocks, 256, 0, stream>>>(
          zbuf, l1W[l], l1b[l], l2W[l], l2b[l], ubuf, N);
    } else {
      mlp_wmma_kernel<7, 8><<<nblocks, 256, 0, stream>>>(
          zbuf, l1W[l], l1b[l], l2W[l], l2b[l], ubuf, N);
    }

    bn_stats_kernel<<<HDIM, 256, 0, stream>>>(ubuf, gam[l], bet[l], scale, shift, N);

    float* hdst = (l == 2) ? node_embs : hbuf;
    bn_apply_kernel<<<(N * HDIM + 255) / 256, 256, 0, stream>>>(ubuf, scale, shift, hdst, N * HDIM);

    hcur = hdst;
    dcur = HDIM;
  }

  pool_kernel<<<(N + 255) / 256, 256, 0, stream>>>(node_embs, batch, graph_emb, N);
  fc_kernel<<<(G * NCLS + 255) / 256, 256, 0, stream>>>(graph_emb, fcW, fcb, out_ptr, G * NCLS);
}